// SetEquiMLP_6708738916907
// MI455X (gfx1250) — compile-verified
//
#include <hip/hip_runtime.h>
#include <hip/hip_bf16.h>

#define BATCH   512
#define NCG     256
#define NAT     1024
#define NKNN    3
#define IN_DIM  (NCG * NKNN)   // 768
#define HID     3072           // LW
#define ODIM    3072           // OUT_DIM
#define KB      64             // K-chunk staged in LDS per TDM transfer

typedef float v2f __attribute__((ext_vector_type(2)));
typedef float v8f __attribute__((ext_vector_type(8)));
typedef unsigned int v4u __attribute__((ext_vector_type(4)));
typedef int v4i __attribute__((ext_vector_type(4)));
typedef int v8i __attribute__((ext_vector_type(8)));

// ---------------------------------------------------------------------------
// TDM: DMA a [rows x 64] f32 tile (row stride = n_total elements) from global
// memory into LDS at byte offset lds_addr.  Descriptor packing per CDNA5 ISA
// §8.3 (group0) / §8.4 (group1); groups 2/3 zero (2D tensor, tile_dim2=0).
// This toolchain exposes the 6-arg builtin:
//   (uint32x4 g0, int32x8 g1, int32x4 g2, int32x4 g3, int32x8 extra, i32 cpol)
// ---------------------------------------------------------------------------
__device__ __forceinline__ void tdm_load_tile(unsigned lds_addr,
                                              const float* gptr,
                                              int rows, int n_total, int k_total) {
    unsigned long long ga = (unsigned long long)(uintptr_t)gptr;
    v4u g0;
    g0.x = 1u;                                        // count=1, user desc
    g0.y = lds_addr;                                  // LDS byte address
    g0.z = (unsigned)(ga & 0xFFFFFFFFu);              // global_addr[31:0]
    g0.w = (unsigned)((ga >> 32) & 0x1FFFFFFu)        // global_addr[56:32]
         | (2u << 30);                                // type = 2 ("image")
    v8i g1;
    g1[0] = 0x20000;                                  // data_size=2 (4 bytes)
    g1[1] = (int)((unsigned)(n_total & 0xFFFF) << 16);            // tensor_dim0[15:0]
    g1[2] = (int)(((unsigned)(n_total >> 16) & 0xFFFFu)           // tensor_dim0[31:16]
                | ((unsigned)(k_total & 0xFFFF) << 16));          // tensor_dim1[15:0]
    g1[3] = (int)(((unsigned)(k_total >> 16) & 0xFFFFu)           // tensor_dim1[31:16]
                | (64u << 16));                                   // tile_dim0 = 64
    g1[4] = rows & 0xFFFF;                            // tile_dim1 (tile_dim2=0)
    g1[5] = n_total;                                  // tensor_dim0_stride[31:0]
    g1[6] = 0;                                        // stride0[47:32], stride1[15:0]
    g1[7] = 0;                                        // stride1[47:16]
    v4i z4 = {0, 0, 0, 0};
    v8i z8 = {0, 0, 0, 0, 0, 0, 0, 0};
    __builtin_amdgcn_tensor_load_to_lds(g0, g1, z4, z4, z8, 0);
}

// ---------------------------------------------------------------------------
// Kernel A: dist_vec[b,n,k,i] = cg_xyz[b,k+1,i] - cg_xyz[b,n,i]
//           dist[b, n*3+k]    = || dist_vec[b,n,k,:] ||
// ---------------------------------------------------------------------------
__global__ __launch_bounds__(256) void dist_kernel(const float* __restrict__ cg_xyz,
                                                   float* __restrict__ dist,
                                                   float* __restrict__ dvec) {
    int gid = blockIdx.x * 256 + threadIdx.x;   // b*NCG + n
    int b = gid >> 8;
    int n = gid & 255;
    const float* base = cg_xyz + (size_t)b * NCG * 3;
    float cx = base[n * 3 + 0];
    float cy = base[n * 3 + 1];
    float cz = base[n * 3 + 2];
    float* dv = dvec + (size_t)gid * 9;
    float* dd = dist + (size_t)b * IN_DIM + n * 3;
#pragma unroll
    for (int k = 0; k < NKNN; ++k) {
        float dx = base[(k + 1) * 3 + 0] - cx;
        float dy = base[(k + 1) * 3 + 1] - cy;
        float dz = base[(k + 1) * 3 + 2] - cz;
        dv[k * 3 + 0] = dx;
        dv[k * 3 + 1] = dy;
        dv[k * 3 + 2] = dz;
        dd[k] = sqrtf(dx * dx + dy * dy + dz * dz);
    }
}

// ---------------------------------------------------------------------------
// WMMA fp32 GEMM with TDM-staged B:  C = op(A[M,K] @ B[K,N] + bias), op=relu?
// Block = 128(M) x 64(N): 8 waves, each owns a 16x64 tile; all waves share the
// same 64-column B stripe, staged in LDS by the Tensor Data Mover in 64x64
// chunks (16 KB), double-buffered.  grid = (N/64, M/128).
//
// Fragment layouts per CDNA5 ISA 7.12.2:
//   A 16x4 (2 VGPR): lanes 0-15 -> (K0,K0+1), lanes 16-31 -> (K0+2,K0+3)
//   B 4x16 (2 VGPR): VGPR0 = rows {K0 | K0+2}, VGPR1 = rows {K0+1 | K0+3}
//   C/D 16x16 (8 VGPR): VGPR r = row (r | r+8), col = lane&15
// ---------------------------------------------------------------------------
__global__ __launch_bounds__(256) void gemm_wmma_f32(const float* __restrict__ A,
                                                     const float* __restrict__ Bm,
                                                     const float* __restrict__ bias,
                                                     float* __restrict__ C,
                                                     int M, int K, int N, int relu) {
    __shared__ float sB[2][KB * 64];

    const int lane  = threadIdx.x & 31;
    const int wave  = threadIdx.x >> 5;
    const int m0    = blockIdx.y * 128 + (wave << 4);
    const int n0    = blockIdx.x << 6;
    const int row   = lane & 15;
    const int col   = lane & 15;
    const int hi    = lane >> 4;        // 0: lanes 0-15, 1: lanes 16-31
    const int khalf = hi << 1;          // 0 or 2

    unsigned lds0 = (unsigned)(uintptr_t)(void*)&sB[0][0];
    unsigned lds1 = (unsigned)(uintptr_t)(void*)&sB[1][0];

    v8f acc0 = {}; v8f acc1 = {}; v8f acc2 = {}; v8f acc3 = {};

    const float* arow = A + (size_t)(m0 + row) * K;
    const int nchunks = K / KB;

    if (wave == 0)
        tdm_load_tile(lds0, Bm + n0, KB, N, K);

    for (int c = 0; c < nchunks; ++c) {
        if (c + 1 < nchunks) {
            if (wave == 0) {
                unsigned dst = ((c + 1) & 1) ? lds1 : lds0;
                tdm_load_tile(dst, Bm + (size_t)(c + 1) * KB * N + n0, KB, N, K);
                __builtin_amdgcn_s_wait_tensorcnt(1);   // chunk c landed
            }
        } else {
            if (wave == 0)
                __builtin_amdgcn_s_wait_tensorcnt(0);   // last chunk landed
        }
        __syncthreads();

        const float* bbuf = &sB[c & 1][0];
        const float* ac   = arow + c * KB;
#pragma unroll 4
        for (int kk = 0; kk < KB; kk += 4) {
            const int kb = kk + khalf;
            v2f a = *(const v2f*)(ac + kb);            // (K=kb, K=kb+1) this lane
            const float* bp0 = bbuf + kb * 64 + col;
            const float* bp1 = bp0 + 64;
            v2f b0; b0.x = bp0[0];  b0.y = bp1[0];
            v2f b1; b1.x = bp0[16]; b1.y = bp1[16];
            v2f b2; b2.x = bp0[32]; b2.y = bp1[32];
            v2f b3; b3.x = bp0[48]; b3.y = bp1[48];
            acc0 = __builtin_amdgcn_wmma_f32_16x16x4_f32(false, a, false, b0, (short)0, acc0, false, false);
            acc1 = __builtin_amdgcn_wmma_f32_16x16x4_f32(false, a, false, b1, (short)0, acc1, false, false);
            acc2 = __builtin_amdgcn_wmma_f32_16x16x4_f32(false, a, false, b2, (short)0, acc2, false, false);
            acc3 = __builtin_amdgcn_wmma_f32_16x16x4_f32(false, a, false, b3, (short)0, acc3, false, false);
        }
        __syncthreads();   // protect LDS buffer before next TDM overwrite
    }

    float bs0 = bias[n0 + 0  + col];
    float bs1 = bias[n0 + 16 + col];
    float bs2 = bias[n0 + 32 + col];
    float bs3 = bias[n0 + 48 + col];
#pragma unroll
    for (int r = 0; r < 8; ++r) {
        int rowg = m0 + r + (hi << 3);
        float* cp = C + (size_t)rowg * N + n0 + col;
        float v0 = acc0[r] + bs0;
        float v1 = acc1[r] + bs1;
        float v2 = acc2[r] + bs2;
        float v3 = acc3[r] + bs3;
        if (relu) {
            v0 = fmaxf(v0, 0.f); v1 = fmaxf(v1, 0.f);
            v2 = fmaxf(v2, 0.f); v3 = fmaxf(v3, 0.f);
        }
        cp[0]  = v0;
        cp[16] = v1;
        cp[32] = v2;
        cp[48] = v3;
    }
}

// ---------------------------------------------------------------------------
// Kernel D: dx_recon[b,a,i] = sum_k coeffs[b, n*12 + k*4 + ch] * dvec[b,n,k,i]
// ---------------------------------------------------------------------------
__global__ __launch_bounds__(256) void dx_recon_kernel(const float* __restrict__ coeffs,
                                                       const float* __restrict__ dvec,
                                                       const int* __restrict__ aidx,
                                                       const int* __restrict__ cidx,
                                                       float* __restrict__ dxr) {
    int gid = blockIdx.x * 256 + threadIdx.x;   // b*NAT + a
    int b = gid >> 10;
    int a = gid & 1023;
    int n  = aidx[a];
    int ch = cidx[a];
    const float* cb = coeffs + (size_t)b * ODIM + n * 12 + ch;   // stride 4 over k
    const float* dv = dvec + ((size_t)b * NCG + n) * 9;
    float o0 = 0.f, o1 = 0.f, o2 = 0.f;
#pragma unroll
    for (int k = 0; k < NKNN; ++k) {
        float c = cb[k * 4];
        o0 += c * dv[k * 3 + 0];
        o1 += c * dv[k * 3 + 1];
        o2 += c * dv[k * 3 + 2];
    }
    float* o = dxr + (size_t)gid * 3;
    o[0] = o0; o[1] = o1; o[2] = o2;
}

// ---------------------------------------------------------------------------
// Kernel E: cg_offset[b,j,i] = sum_a assign_norm[b,a,j] * dx_recon[b,a,i]
// one block per batch; assign_norm reads coalesced; dx chunk staged in LDS.
// ---------------------------------------------------------------------------
__global__ __launch_bounds__(256) void cg_offset_kernel(const float* __restrict__ an,
                                                        const float* __restrict__ dxr,
                                                        float* __restrict__ cgo) {
    __shared__ float s_dx[256 * 3];
    int b = blockIdx.x;
    int j = threadIdx.x;
    const float* anb = an + (size_t)b * NAT * NCG;
    const float* dxb = dxr + (size_t)b * NAT * 3;
    float acc0 = 0.f, acc1 = 0.f, acc2 = 0.f;
    for (int a0 = 0; a0 < NAT; a0 += 256) {
        for (int t = threadIdx.x; t < 768; t += 256)
            s_dx[t] = dxb[a0 * 3 + t];
        __syncthreads();
        for (int a = 0; a < 256; ++a) {
            float w = anb[(size_t)(a0 + a) * NCG + j];   // coalesced
            acc0 += w * s_dx[a * 3 + 0];                 // broadcast
            acc1 += w * s_dx[a * 3 + 1];
            acc2 += w * s_dx[a * 3 + 2];
        }
        __syncthreads();
    }
    float* o = cgo + ((size_t)b * NCG + j) * 3;
    o[0] = acc0; o[1] = acc1; o[2] = acc2;
}

// ---------------------------------------------------------------------------
// Kernel F: xyz_recon[b,a,i] = cg_xyz[b,n,i] - cg_offset[b,n,i] + dx_recon[b,a,i]
// ---------------------------------------------------------------------------
__global__ __launch_bounds__(256) void recon_kernel(const float* __restrict__ cg_xyz,
                                                    const float* __restrict__ cgo,
                                                    const float* __restrict__ dxr,
                                                    const int* __restrict__ aidx,
                                                    float* __restrict__ out) {
    int gid = blockIdx.x * 256 + threadIdx.x;   // b*NAT + a
    int b = gid >> 10;
    int a = gid & 1023;
    int n = aidx[a];
    const float* cgx = cg_xyz + ((size_t)b * NCG + n) * 3;
    const float* co  = cgo + ((size_t)b * NCG + n) * 3;
    const float* dx  = dxr + (size_t)gid * 3;
    float* o = out + (size_t)gid * 3;
#pragma unroll
    for (int i = 0; i < 3; ++i)
        o[i] = cgx[i] - co[i] + dx[i];
}

// ---------------------------------------------------------------------------
extern "C" void kernel_launch(void* const* d_in, const int* in_sizes, int n_in,
                              void* d_out, int out_size, void* d_ws, size_t ws_size,
                              hipStream_t stream) {
    const float* xyz         = (const float*)d_in[0];
    const float* cg_xyz      = (const float*)d_in[1];
    const float* assign_norm = (const float*)d_in[2];
    const float* soft_assign = (const float*)d_in[3];
    const float* W1          = (const float*)d_in[4];
    const float* b1          = (const float*)d_in[5];
    const float* W2          = (const float*)d_in[6];
    const float* b2          = (const float*)d_in[7];
    const int*   assign_idx  = (const int*)d_in[8];
    const int*   channel_idx = (const int*)d_in[9];

    float* ws = (float*)d_ws;
    float* w_dist   = ws;                                  // 512*768      = 393216
    float* w_dvec   = w_dist + (size_t)BATCH * IN_DIM;     // 512*256*9    = 1179648
    float* w_h      = w_dvec + (size_t)BATCH * NCG * 9;    // 512*3072     = 1572864
    float* w_coeff  = w_h + (size_t)BATCH * HID;           // 512*3072     = 1572864
    float* w_dxr    = w_coeff + (size_t)BATCH * ODIM;      // 512*1024*3   = 1572864
    float* w_cgo    = w_dxr + (size_t)BATCH * NAT * 3;     // 512*256*3    = 393216
    // total ~26.7 MB of d_ws

    // ---- passthrough outputs: soft_assign then xyz, via async d2d copies ----
    const size_t soft_elems = (size_t)BATCH * NAT * NCG;   // 134217728
    const size_t xyz_elems  = (size_t)BATCH * NAT * 3;     // 1572864
    float* out_soft  = (float*)d_out;
    float* out_xyz   = out_soft + soft_elems;
    float* out_recon = out_xyz + xyz_elems;
    (void)hipMemcpyAsync(out_soft, soft_assign, soft_elems * sizeof(float),
                         hipMemcpyDeviceToDevice, stream);
    (void)hipMemcpyAsync(out_xyz, xyz, xyz_elems * sizeof(float),
                         hipMemcpyDeviceToDevice, stream);

    // ---- A: distances ----
    dist_kernel<<<(BATCH * NCG) / 256, 256, 0, stream>>>(cg_xyz, w_dist, w_dvec);

    // ---- B: h = relu(dist @ W1 + b1), (512x768)@(768x3072) ----
    {
        dim3 grid(HID / 64, BATCH / 128);
        gemm_wmma_f32<<<grid, 256, 0, stream>>>(w_dist, W1, b1, w_h,
                                                BATCH, IN_DIM, HID, 1);
    }
    // ---- C: coeffs = h @ W2 + b2, (512x3072)@(3072x3072) ----
    {
        dim3 grid(ODIM / 64, BATCH / 128);
        gemm_wmma_f32<<<grid, 256, 0, stream>>>(w_h, W2, b2, w_coeff,
                                                BATCH, HID, ODIM, 0);
    }

    // ---- D: dx_recon gather-contract ----
    dx_recon_kernel<<<(BATCH * NAT) / 256, 256, 0, stream>>>(w_coeff, w_dvec,
                                                             assign_idx, channel_idx,
                                                             w_dxr);

    // ---- E: cg_offset = assign_norm^T @ dx_recon per batch ----
    cg_offset_kernel<<<BATCH, 256, 0, stream>>>(assign_norm, w_dxr, w_cgo);

    // ---- F: xyz_recon ----
    recon_kernel<<<(BATCH * NAT) / 256, 256, 0, stream>>>(cg_xyz, w_cgo, w_dxr,
                                                          assign_idx, out_recon);
}